// RSRConv_29386166239302
// MI455X (gfx1250) — compile-verified
//
#include <hip/hip_runtime.h>

#define NN 50000
#define NE 600000
#define DD 128
#define NT 8

typedef __attribute__((ext_vector_type(2))) float v2f;
typedef __attribute__((ext_vector_type(8))) float v8f;

// ---------------------------------------------------------------------------
// Phase 0: init output + segment buffers (harness does not re-zero between
// replays, so every launch must initialize its own accumulators).
// ---------------------------------------------------------------------------
__global__ void rsr_init_kernel(float* __restrict__ out,
                                float* __restrict__ seg_max,
                                float* __restrict__ seg_sum,
                                int nd, int n) {
    int i = blockIdx.x * blockDim.x + threadIdx.x;
    if (i < nd) out[i] = 0.0f;
    if (i < n) {
        seg_max[i] = -__builtin_inff();
        seg_sum[i] = 0.0f;
    }
}

// ---------------------------------------------------------------------------
// Phase 1: el/er node scores via V_WMMA_F32_16X16X4_F32.
//   A (16x4 f32)  : 16 nodes x 4-feature chunk.
//      lane layout: lanes 0-15 -> M=lane, K={0,1}; lanes 16-31 -> M=lane-16, K={2,3}
//   B (4x16 f32)  : column 0 = head_w chunk, column 1 = tail_w chunk, rest 0.
//   C (16x16 f32) : column 0 accumulates el, column 1 accumulates er.
//
// B is built branch-free: one pointer select outside the loop + a per-lane
// multiplicative mask (avoids per-iteration EXEC-mask branching that round 1
// codegen produced). K-loop fully unrolled: 64 global_load_b64 pipelined
// against a 32-deep f32 WMMA accumulation chain. One wave per 16-node tile.
// ---------------------------------------------------------------------------
__global__ void rsr_scores_kernel(const float* __restrict__ nfeat,
                                  const float* __restrict__ head_w,
                                  const float* __restrict__ head_b,
                                  const float* __restrict__ tail_w,
                                  const float* __restrict__ tail_b,
                                  float* __restrict__ el,
                                  float* __restrict__ er) {
    const int wave   = threadIdx.x >> 5;
    const int tile   = blockIdx.x * (blockDim.x >> 5) + wave;
    if (tile * 16 >= NN) return;                 // whole-wave guard (EXEC all-1 inside)

    const int lane   = threadIdx.x & 31;
    const int lane15 = lane & 15;                // A: row M; B/C: column N
    const int khalf  = lane >> 4;                // which K pair this lane holds

    const float* __restrict__ arow = nfeat + (size_t)(tile * 16 + lane15) * DD;

    // Branch-free B source: column 0 reads head_w, column 1 reads tail_w,
    // all other columns load head_w but multiply by 0.
    const float* __restrict__ wbase = (lane15 == 1) ? tail_w : head_w;
    const float wmask = (lane15 < 2) ? 1.0f : 0.0f;

    v8f c = {0.f, 0.f, 0.f, 0.f, 0.f, 0.f, 0.f, 0.f};

#pragma unroll
    for (int kb = 0; kb < DD; kb += 4) {
        const int k0 = kb + khalf * 2;
        const v2f a = *(const v2f*)(arow + k0);   // 8B-aligned global_load_b64
        const v2f w = *(const v2f*)(wbase + k0);  // 8B-aligned global_load_b64
        v2f b;
        b.x = w.x * wmask;
        b.y = w.y * wmask;
        c = __builtin_amdgcn_wmma_f32_16x16x4_f32(
                /*neg_a=*/false, a, /*neg_b=*/false, b,
                /*c_mod=*/(short)0, c, /*reuse_a=*/false, /*reuse_b=*/false);
    }

    // C layout: VGPR i holds M = i + 8*(lane>=16), N = lane15.
    // Column 0 (lanes 0 and 16) -> el ; column 1 (lanes 1 and 17) -> er.
    if (lane15 < 2) {
        float* __restrict__ dstp = (lane15 == 0) ? el : er;
        const float bias = (lane15 == 0) ? head_b[0] : tail_b[0];
        const int base = tile * 16 + khalf * 8;
        dstp[base + 0] = c[0] + bias;
        dstp[base + 1] = c[1] + bias;
        dstp[base + 2] = c[2] + bias;
        dstp[base + 3] = c[3] + bias;
        dstp[base + 4] = c[4] + bias;
        dstp[base + 5] = c[5] + bias;
        dstp[base + 6] = c[6] + bias;
        dstp[base + 7] = c[7] + bias;
    }
}

// ---------------------------------------------------------------------------
// Float atomic max via signed/unsigned int ordering trick
// (lowers to native global_atomic_max_i32 / global_atomic_min_u32).
// seg_max must be initialized to -inf (0xFF800000).
// ---------------------------------------------------------------------------
__device__ __forceinline__ void atomicMaxF32(float* addr, float val) {
    if (val >= 0.0f) {
        atomicMax((int*)addr, __float_as_int(val));
    } else {
        atomicMin((unsigned int*)addr, __float_as_uint(val));
    }
}

// ---------------------------------------------------------------------------
// Phase 2: per-edge logits + segment max over dst.
// ---------------------------------------------------------------------------
__global__ void rsr_logits_kernel(const int* __restrict__ src,
                                  const int* __restrict__ dst,
                                  const int* __restrict__ etype,
                                  const float* __restrict__ el,
                                  const float* __restrict__ er,
                                  const float* __restrict__ rel_w,
                                  float* __restrict__ logits,
                                  float* __restrict__ seg_max) {
    int e = blockIdx.x * blockDim.x + threadIdx.x;
    if (e >= NE) return;
    const int s = src[e];
    const int d = dst[e];
    const int t = etype[e];
    const float l = el[s] + er[d] + rel_w[t];
    logits[e] = l;
    atomicMaxF32(seg_max + d, l);
}

// ---------------------------------------------------------------------------
// Phase 3: ex = exp(logit - seg_max[dst]); segment sum over dst.
// Overwrites the logits slot with ex.
// ---------------------------------------------------------------------------
__global__ void rsr_exp_kernel(const int* __restrict__ dst,
                               const float* __restrict__ seg_max,
                               float* __restrict__ exbuf,
                               float* __restrict__ seg_sum) {
    int e = blockIdx.x * blockDim.x + threadIdx.x;
    if (e >= NE) return;
    const int d = dst[e];
    const float ex = __expf(exbuf[e] - seg_max[d]);
    exbuf[e] = ex;
    atomicAdd(seg_sum + d, ex);
}

// ---------------------------------------------------------------------------
// Phase 4: out[dst] += (ex/seg_sum[dst]) * nfeat[src].
// One wave32 per edge; each lane owns a float4 slice of the 128-wide row
// (global_load_b128 gather from L2-resident nfeat, 4 f32 atomics to L2).
// nfeat (25.6 MB) and out (25.6 MB) both fit in the 192 MB L2, so the
// 600K x 512B gathers and 76.8M atomic adds resolve at L2, not HBM.
// ---------------------------------------------------------------------------
__global__ void rsr_aggregate_kernel(const int* __restrict__ src,
                                     const int* __restrict__ dst,
                                     const float* __restrict__ exbuf,
                                     const float* __restrict__ seg_sum,
                                     const float* __restrict__ nfeat,
                                     float* __restrict__ out) {
    const int e = blockIdx.x * (blockDim.x >> 5) + (threadIdx.x >> 5);
    if (e >= NE) return;
    const int lane = threadIdx.x & 31;
    const int s = src[e];
    const int d = dst[e];
    const float a = exbuf[e] / seg_sum[d];

    const float4 v = ((const float4*)(nfeat + (size_t)s * DD))[lane];
    float* __restrict__ orow = out + (size_t)d * DD + lane * 4;
    atomicAdd(orow + 0, a * v.x);
    atomicAdd(orow + 1, a * v.y);
    atomicAdd(orow + 2, a * v.z);
    atomicAdd(orow + 3, a * v.w);
}

// ---------------------------------------------------------------------------
extern "C" void kernel_launch(void* const* d_in, const int* in_sizes, int n_in,
                              void* d_out, int out_size, void* d_ws, size_t ws_size,
                              hipStream_t stream) {
    (void)in_sizes; (void)n_in; (void)out_size; (void)ws_size;

    const float* nfeat  = (const float*)d_in[0];
    const float* head_w = (const float*)d_in[1];
    const float* head_b = (const float*)d_in[2];
    const float* tail_w = (const float*)d_in[3];
    const float* tail_b = (const float*)d_in[4];
    const float* rel_w  = (const float*)d_in[5];
    const int*   src    = (const int*)d_in[6];
    const int*   dst    = (const int*)d_in[7];
    const int*   etype  = (const int*)d_in[8];

    float* out = (float*)d_out;
    float* ws  = (float*)d_ws;
    float* el      = ws;                 // [NN]
    float* er      = ws + 1 * NN;        // [NN]
    float* seg_max = ws + 2 * NN;        // [NN]
    float* seg_sum = ws + 3 * NN;        // [NN]
    float* exbuf   = ws + 4 * NN;        // [NE] logits, then ex

    const int nd = NN * DD;
    rsr_init_kernel<<<(nd + 255) / 256, 256, 0, stream>>>(out, seg_max, seg_sum, nd, NN);

    const int tiles = (NN + 15) / 16;    // 3125 (divides exactly: 50000/16)
    const int wpb   = 8;                 // waves (tiles) per 256-thread block
    rsr_scores_kernel<<<(tiles + wpb - 1) / wpb, 32 * wpb, 0, stream>>>(
        nfeat, head_w, head_b, tail_w, tail_b, el, er);

    rsr_logits_kernel<<<(NE + 255) / 256, 256, 0, stream>>>(
        src, dst, etype, el, er, rel_w, exbuf, seg_max);

    rsr_exp_kernel<<<(NE + 255) / 256, 256, 0, stream>>>(
        dst, seg_max, exbuf, seg_sum);

    rsr_aggregate_kernel<<<(NE + wpb - 1) / wpb, 32 * wpb, 0, stream>>>(
        src, dst, exbuf, seg_sum, nfeat, out);
}